// ClassicalGCN_77077483094916
// MI455X (gfx1250) — compile-verified
//
#include <hip/hip_runtime.h>
#include <hip/hip_bf16.h>

typedef __attribute__((ext_vector_type(2))) float v2f;
typedef __attribute__((ext_vector_type(8))) float v8f;

#define IN_DIM 128
#define HID    64

#if __has_builtin(__builtin_amdgcn_tanhf)
__device__ __forceinline__ float fast_tanh(float v) { return __builtin_amdgcn_tanhf(v); }
#else
__device__ __forceinline__ float fast_tanh(float v) { return tanhf(v); }
#endif

// ---------------------------------------------------------------------------
// Stage 1: s[j] = tanh(x[j] @ W1 + b1) . W2      (one scalar per node)
// One wave computes a 16-node tile via V_WMMA_F32_16X16X4_F32.
// A (16x4 f32): lanes 0-15 -> M=lane, K={k0,k0+1}; lanes 16-31 -> K={k0+2,k0+3}
// B (4x16 f32): lanes 0-15 -> N=lane, K={k0,k0+1}; lanes 16-31 -> K={k0+2,k0+3}
// C/D (16x16 f32, 8 VGPRs): vgpr r -> M = r + 8*(lane>=16), N = lane%16
// ---------------------------------------------------------------------------
__global__ __launch_bounds__(128, 4)
void gcn_stage1_wmma(const float* __restrict__ x,
                     const float* __restrict__ W1,
                     const float* __restrict__ b1,
                     const float* __restrict__ W2,
                     float* __restrict__ s,
                     int nTiles)
{
    const int lane    = threadIdx.x & 31;
    const int wave    = threadIdx.x >> 5;
    const int tile    = blockIdx.x * 4 + wave;     // wave-uniform
    if (tile >= nTiles) return;                    // uniform exit: EXEC stays all-1s
    const int node0   = tile * 16;
    const int halfsel = lane >> 4;                 // 0 | 1
    const int l15     = lane & 15;

    const float* xrow = x + (size_t)(node0 + l15) * IN_DIM + halfsel * 2;

    v8f acc0 = {}, acc1 = {}, acc2 = {}, acc3 = {};

    for (int k0 = 0; k0 < IN_DIM; k0 += 4) {
        v2f a = *(const v2f*)(xrow + k0);

        const int    krow  = k0 + halfsel * 2;
        const float* wbase = W1 + (size_t)krow * HID + l15;
        v2f bf0, bf1, bf2, bf3;
        bf0.x = wbase[0];   bf0.y = wbase[HID + 0];
        bf1.x = wbase[16];  bf1.y = wbase[HID + 16];
        bf2.x = wbase[32];  bf2.y = wbase[HID + 32];
        bf3.x = wbase[48];  bf3.y = wbase[HID + 48];

        acc0 = __builtin_amdgcn_wmma_f32_16x16x4_f32(false, a, false, bf0, (short)0, acc0, false, false);
        acc1 = __builtin_amdgcn_wmma_f32_16x16x4_f32(false, a, false, bf1, (short)0, acc1, false, false);
        acc2 = __builtin_amdgcn_wmma_f32_16x16x4_f32(false, a, false, bf2, (short)0, acc2, false, false);
        acc3 = __builtin_amdgcn_wmma_f32_16x16x4_f32(false, a, false, bf3, (short)0, acc3, false, false);
    }

    // Epilogue: bias + tanh + dot with W2, per-lane partial over this lane's 4 columns.
    v8f accs[4] = {acc0, acc1, acc2, acc3};
    float part[8];
    #pragma unroll
    for (int r = 0; r < 8; ++r) part[r] = 0.0f;

    #pragma unroll
    for (int t = 0; t < 4; ++t) {
        const int   n    = t * 16 + l15;
        const float bias = b1[n];
        const float w2v  = W2[n];
        #pragma unroll
        for (int r = 0; r < 8; ++r) {
            float h = fast_tanh(accs[t][r] + bias);
            part[r] += h * w2v;
        }
    }

    // Reduce across the 16 lanes that share each row group (xor tree stays in-half).
    #pragma unroll
    for (int r = 0; r < 8; ++r) {
        float v = part[r];
        v += __shfl_xor(v, 1, 32);
        v += __shfl_xor(v, 2, 32);
        v += __shfl_xor(v, 4, 32);
        v += __shfl_xor(v, 8, 32);
        part[r] = v;
    }

    if (l15 == 0) {
        const int mbase = node0 + halfsel * 8;     // lane 0 -> rows 0..7, lane 16 -> rows 8..15
        #pragma unroll
        for (int r = 0; r < 8; ++r) s[mbase + r] = part[r];
    }
}

// ---------------------------------------------------------------------------
// Stage 2a: out[i] = b2
// ---------------------------------------------------------------------------
__global__ __launch_bounds__(256)
void gcn_init_out(float* __restrict__ out, const float* __restrict__ b2, int n)
{
    int i = blockIdx.x * blockDim.x + threadIdx.x;
    if (i < n) out[i] = b2[0];
}

// ---------------------------------------------------------------------------
// Stage 2b: out[row[e]] += val[e] * s[col[e]]   (s is L2-resident, 200 KB)
// 4 edges per iteration via b128 loads on the 19.2 MB edge stream.
// ---------------------------------------------------------------------------
__global__ __launch_bounds__(256)
void gcn_edge_scatter(const int* __restrict__ erow,
                      const int* __restrict__ ecol,
                      const float* __restrict__ eval,
                      const float* __restrict__ s,
                      float* __restrict__ out,
                      int E)
{
    const int stride = gridDim.x * blockDim.x;
    const int tid    = blockIdx.x * blockDim.x + threadIdx.x;
    const int E4     = E >> 2;

    const int4*   erow4 = (const int4*)erow;
    const int4*   ecol4 = (const int4*)ecol;
    const float4* eval4 = (const float4*)eval;

    for (int q = tid; q < E4; q += stride) {
        int4   r = erow4[q];
        int4   c = ecol4[q];
        float4 v = eval4[q];
        __hip_atomic_fetch_add(&out[r.x], v.x * s[c.x], __ATOMIC_RELAXED, __HIP_MEMORY_SCOPE_AGENT);
        __hip_atomic_fetch_add(&out[r.y], v.y * s[c.y], __ATOMIC_RELAXED, __HIP_MEMORY_SCOPE_AGENT);
        __hip_atomic_fetch_add(&out[r.z], v.z * s[c.z], __ATOMIC_RELAXED, __HIP_MEMORY_SCOPE_AGENT);
        __hip_atomic_fetch_add(&out[r.w], v.w * s[c.w], __ATOMIC_RELAXED, __HIP_MEMORY_SCOPE_AGENT);
    }
    // tail (E % 4 edges)
    for (int e = E4 * 4 + tid; e < E; e += stride) {
        __hip_atomic_fetch_add(&out[erow[e]], eval[e] * s[ecol[e]],
                               __ATOMIC_RELAXED, __HIP_MEMORY_SCOPE_AGENT);
    }
}

// ---------------------------------------------------------------------------
extern "C" void kernel_launch(void* const* d_in, const int* in_sizes, int n_in,
                              void* d_out, int out_size, void* d_ws, size_t ws_size,
                              hipStream_t stream)
{
    const float* x    = (const float*)d_in[0];
    const int*   erow = (const int*)  d_in[1];
    const int*   ecol = (const int*)  d_in[2];
    const float* eval = (const float*)d_in[3];
    const float* W1   = (const float*)d_in[4];
    const float* b1   = (const float*)d_in[5];
    const float* W2   = (const float*)d_in[6];
    const float* b2   = (const float*)d_in[7];
    float*       out  = (float*)d_out;

    const int N = in_sizes[0] / IN_DIM;   // 50000
    const int E = in_sizes[1];            // 1600000

    float* s = (float*)d_ws;              // N floats of scratch (node scalars)

    const int nTiles  = (N + 15) / 16;          // 3125 (exact)
    const int blocks1 = (nTiles + 3) / 4;       // 4 waves (tiles) per 128-thread block

    gcn_stage1_wmma<<<blocks1, 128, 0, stream>>>(x, W1, b1, W2, s, nTiles);
    gcn_init_out<<<(N + 255) / 256, 256, 0, stream>>>(out, b2, N);
    gcn_edge_scatter<<<1024, 256, 0, stream>>>(erow, ecol, eval, s, out, E);
}